// TransGQARopeIndexer_46480136077927
// MI455X (gfx1250) — compile-verified
//
#include <hip/hip_runtime.h>
#include <hip/hip_bf16.h>
#include <math.h>

// Problem constants (from reference): N=16384, H=32, G=8, D=128, FF=2, R=4
// GROUPS = 4, NBAND = 32.
#define NTOK   16384
#define NHEAD  32
#define NKV    8
#define DIM    128
#define NGRP   4
#define NBAND  32

typedef __attribute__((ext_vector_type(2))) float v2f;
typedef __attribute__((ext_vector_type(8))) float v8f;

// ---------------------------------------------------------------------------
// weights kernel: per 16-token tile, 8 kv-group GEMMs [16x128]x[128x16] via
// V_WMMA_F32_16X16X4_F32 (32 k-steps each), square-accumulate, 4-lane shuffle
// reduce, sqrt, store. B (reshaped v_transform) staged in LDS in WMMA layout.
// Block = 256 threads = 8 waves = 128 tokens.
// ---------------------------------------------------------------------------
__global__ __launch_bounds__(256) void weights_kernel(
    const float* __restrict__ v,    // [N, NKV, DIM]
    const float* __restrict__ vt,   // [NHEAD, DIM, 4]
    float* __restrict__ outw)       // [N, NGRP]
{
    // Bl[g][d>>1][col][d&1] = vt[g*4 + (col>>2)][d][col&3]   (64 KB)
    __shared__ float Bl[NKV * 64 * 16 * 2];

    // Cooperative staging: coalesced global read, scattered LDS write.
    for (int idx = threadIdx.x; idx < NHEAD * DIM * 4; idx += 256) {
        float val = vt[idx];
        int h = idx >> 9;             // head
        int d = (idx >> 2) & 127;     // k index
        int r = idx & 3;              // rank
        int g = h >> 2;
        int col = ((h & 3) << 2) | r; // (grp, r) column 0..15
        Bl[((g * 64 + (d >> 1)) * 16 + col) * 2 + (d & 1)] = val;
    }
    __syncthreads();

    const int lane = threadIdx.x & 31;
    const int wave = threadIdx.x >> 5;
    const int hh   = lane >> 4;            // 0: K pair {k0,k0+1}, 1: {k0+2,k0+3}
    const int koff = hh << 1;
    const long tok0 = (long)blockIdx.x * 128 + wave * 16;

    // A fragment base: v row of this lane's token, shifted by koff.
    const float* vrow  = v + (tok0 + (lane & 15)) * (NKV * DIM) + koff;
    // B fragment base inside LDS (per-(g,step) offset = g*2048 + s*64 floats).
    const float* bbase = &Bl[(size_t)hh * 32 + (size_t)(lane & 15) * 2];

    float sq[8] = {0.f, 0.f, 0.f, 0.f, 0.f, 0.f, 0.f, 0.f};

    for (int g = 0; g < NKV; ++g) {
        v8f c = {0.f, 0.f, 0.f, 0.f, 0.f, 0.f, 0.f, 0.f};
#pragma unroll
        for (int s = 0; s < 32; ++s) {
            v2f a = *(const v2f*)(vrow + g * DIM + s * 4);
            v2f b = *(const v2f*)(bbase + g * 2048 + s * 64);
            c = __builtin_amdgcn_wmma_f32_16x16x4_f32(
                    false, a, false, b, (short)0, c, false, false);
        }
#pragma unroll
        for (int i = 0; i < 8; ++i) sq[i] += c[i] * c[i];
    }

    // C layout: vgpr i -> (M = i + 8*hh, Ncol = lane&15). Reduce r (4 adjacent
    // columns) with xor-shuffles, lane col%4==0 holds the (m, grp) sum.
#pragma unroll
    for (int i = 0; i < 8; ++i) {
        float s = sq[i];
        s += __shfl_xor(s, 1, 32);
        s += __shfl_xor(s, 2, 32);
        if ((lane & 3) == 0) {
            int m   = i + (hh << 3);
            int grp = (lane >> 2) & 3;
            outw[(tok0 + m) * NGRP + grp] = sqrtf(s);
        }
    }
}

// ---------------------------------------------------------------------------
// index_q kernel: one block per token (256 threads). Thread = (grp, half, nb).
// 16 gathered q reads (each q element consumed exactly once) + 16 FMAs against
// L0-resident qk_index_proj, coalesced float2 store.
// ---------------------------------------------------------------------------
__global__ __launch_bounds__(256) void idxq_kernel(
    const float* __restrict__ q,   // [N, NHEAD, DIM]
    const float* __restrict__ P,   // [NBAND, 16, 2]
    float* __restrict__ outq)      // [N, NGRP, DIM]
{
    const int b    = blockIdx.x;
    const int sub  = threadIdx.x;
    const int grp  = sub >> 6;
    const int half = (sub >> 5) & 1;
    const int nb   = sub & 31;

    const float* qb = q + (size_t)b * (NHEAD * DIM) + half * 64;
    const float* Pn = P + nb * 32;

    float a0 = 0.f, a1 = 0.f;
#pragma unroll
    for (int f = 0; f < 16; ++f) {
        int g = f & 7;
        int d = ((nb & 15) << 2) | ((f >> 3) << 1) | (nb >> 4);
        float x = qb[(g * NGRP + grp) * DIM + d];
        v2f p = *(const v2f*)(Pn + f * 2);
        a0 += x * p.x;
        a1 += x * p.y;
    }
    float2 r; r.x = a0; r.y = a1;
    *(float2*)(outq + ((size_t)b * NGRP + grp) * DIM + half * 64 + nb * 2) = r;
}

// ---------------------------------------------------------------------------
// index_k kernel: thread = (token, half, nb); 64 threads per token.
// ---------------------------------------------------------------------------
__global__ __launch_bounds__(256) void idxk_kernel(
    const float* __restrict__ k,   // [N, NKV, DIM]
    const float* __restrict__ P,   // [NBAND, 16, 2]
    float* __restrict__ outk)      // [N, DIM]
{
    const int tid  = blockIdx.x * 256 + threadIdx.x;
    const int b    = tid >> 6;
    const int sub  = tid & 63;
    const int half = sub >> 5;
    const int nb   = sub & 31;

    const float* kb = k + (size_t)b * (NKV * DIM) + half * 64;
    const float* Pn = P + nb * 32;

    float a0 = 0.f, a1 = 0.f;
#pragma unroll
    for (int f = 0; f < 16; ++f) {
        int g = f & 7;
        int d = ((nb & 15) << 2) | ((f >> 3) << 1) | (nb >> 4);
        float x = kb[g * DIM + d];
        v2f p = *(const v2f*)(Pn + f * 2);
        a0 += x * p.x;
        a1 += x * p.y;
    }
    float2 r; r.x = a0; r.y = a1;
    *(float2*)(outk + (size_t)b * DIM + half * 64 + nb * 2) = r;
}

// ---------------------------------------------------------------------------
extern "C" void kernel_launch(void* const* d_in, const int* in_sizes, int n_in,
                              void* d_out, int out_size, void* d_ws, size_t ws_size,
                              hipStream_t stream) {
    const float* q  = (const float*)d_in[0];   // [N, 32, 128]
    const float* k  = (const float*)d_in[1];   // [N, 8, 128]
    const float* v  = (const float*)d_in[2];   // [N, 8, 128]
    const float* P  = (const float*)d_in[3];   // [32, 16, 2]
    const float* vt = (const float*)d_in[4];   // [32, 128, 4]

    float* out   = (float*)d_out;
    float* out_q = out;                                       // N*4*128
    float* out_k = out + (size_t)NTOK * NGRP * DIM;           // N*128
    float* out_w = out_k + (size_t)NTOK * DIM;                // N*4

    idxq_kernel   <<<NTOK,            256, 0, stream>>>(q, P, out_q);
    idxk_kernel   <<<(NTOK * 64)/256, 256, 0, stream>>>(k, P, out_k);
    weights_kernel<<<NTOK / 128,      256, 0, stream>>>(v, vt, out_w);
}